// NLayerGCN_12601434046863
// MI455X (gfx1250) — compile-verified
//
#include <hip/hip_runtime.h>
#include <hip/hip_bf16.h>

// ---------------------------------------------------------------------------
// 3-layer GCN on MI455X (gfx1250, wave32).
//   per layer:  H = X @ W            (FP32 WMMA 16x16x4, LDS-staged A tile)
//               out = H*dinv^2 + b   (self loop + bias, float4)
//               out[dst] += H[src]*dinv[src]*dinv[dst]  (edge scatter, L2 atomics)
// ---------------------------------------------------------------------------

#define GCN_N   50000
#define GCN_E   500000
#define D_IN    128
#define D_HID   128
#define D_OUT   64

typedef __attribute__((ext_vector_type(2))) float v2f;
typedef __attribute__((ext_vector_type(8))) float v8f;
typedef __attribute__((ext_vector_type(4))) int   v4i;

#if __has_builtin(__builtin_amdgcn_global_load_async_to_lds_b128) && \
    __has_builtin(__builtin_amdgcn_s_wait_asynccnt)
#define GCN_USE_ASYNC_LDS 1
typedef __attribute__((address_space(1))) v4i* v4i_glob;
typedef __attribute__((address_space(3))) v4i* v4i_lds;
#endif

// ----------------------- degree / normalization ----------------------------

__global__ void gcn_init_deg(float* __restrict__ deg, int n) {
    int i = blockIdx.x * blockDim.x + threadIdx.x;
    if (i < n) deg[i] = 1.0f;               // self loop contributes 1
}

__global__ void gcn_accum_deg(const long long* __restrict__ ei,
                              float* __restrict__ deg, int e) {
    int i = blockIdx.x * blockDim.x + threadIdx.x;
    if (i < e) {
        int d = (int)ei[GCN_E + i];         // dst row of edge_index
        atomicAdd(&deg[d], 1.0f);
    }
}

__global__ void gcn_finalize_dinv(float* __restrict__ deg, int n) {
    int i = blockIdx.x * blockDim.x + threadIdx.x;
    if (i < n) {
        float d = deg[i];
        deg[i] = (d > 0.0f) ? rsqrtf(d) : 0.0f;   // in-place -> dinv
    }
}

// ----------------------------- WMMA GEMM -----------------------------------
// H[M x NOUT] = X[M x K] @ W[K x NOUT], row-major fp32, K/NOUT compile-time.
// One block per 16-row M tile; NOUT/16 waves, one 16x16 output tile each.
// The 16xK X tile is staged in LDS once (async-to-LDS when available) and
// shared by all waves; W operand loads use immediate offsets.
template<int K, int NOUT>
__global__ __launch_bounds__(256)
void gcn_gemm_wmma(const float* __restrict__ X, const float* __restrict__ W,
                   float* __restrict__ H) {
    constexpr int WAVES = NOUT / 16;
    constexpr int TPB   = WAVES * 32;
    __shared__ float ldsX[16 * K];                       // 16 rows x K

    const int tm = blockIdx.x;
    const float* __restrict__ Xt = X + (size_t)tm * 16 * K;

    // ---- stage X tile into LDS (float4 granularity, coalesced) ----
    constexpr int VECS = 16 * K / 4;                     // float4 count
#ifdef GCN_USE_ASYNC_LDS
    #pragma unroll
    for (int v = threadIdx.x; v < VECS; v += TPB) {
        __builtin_amdgcn_global_load_async_to_lds_b128(
            (v4i_glob)(Xt + (size_t)v * 4),
            (v4i_lds)(ldsX + v * 4),
            /*offset=*/0, /*cpol=*/0);
    }
    __builtin_amdgcn_s_wait_asynccnt(0);
    __syncthreads();
#else
    #pragma unroll
    for (int v = threadIdx.x; v < VECS; v += TPB) {
        ((float4*)ldsX)[v] = ((const float4*)Xt)[v];
    }
    __syncthreads();
#endif

    const int wave = threadIdx.x >> 5;
    const int lane = threadIdx.x & 31;
    const int half = lane >> 4;              // 0: lanes 0-15, 1: lanes 16-31
    const int l16  = lane & 15;
    const int col  = wave * 16 + l16;

    // A 16x4 layout: lower half-lanes carry K={0,1} in v0/v1, upper K={2,3}.
    const float* __restrict__ Arow = ldsX + l16 * K;

    v8f acc = {};
    #pragma unroll
    for (int k = 0; k < K; k += 4) {
        const int ka = k + half * 2;
        v2f a, b;
        a.x = Arow[ka + 0];                  // ds_load (8B aligned pair)
        a.y = Arow[ka + 1];
        b.x = W[(ka + 0) * NOUT + col];      // immediate-offset global loads
        b.y = W[(ka + 1) * NOUT + col];
        acc = __builtin_amdgcn_wmma_f32_16x16x4_f32(
                  false, a, false, b, (short)0, acc, false, false);
    }

    // C/D layout: VGPR r -> row (tm*16 + r + half*8), col (wave*16 + l16).
    float* __restrict__ Hout = H + (size_t)(tm * 16) * NOUT + col;
    #pragma unroll
    for (int r = 0; r < 8; ++r) {
        Hout[(r + half * 8) * NOUT] = acc[r];
    }
}

// -------------------- self loop + bias initialization ----------------------
// out[i][f..f+3] = H[i][f..f+3] * dinv[i]^2 + b[f..f+3]
template<int F>
__global__ void gcn_self_bias(const float* __restrict__ H,
                              const float* __restrict__ dinv,
                              const float* __restrict__ b,
                              float* __restrict__ out, int n) {
    constexpr int Q = F / 4;
    int idx = blockIdx.x * blockDim.x + threadIdx.x;     // over n*F/4
    if (idx < n * Q) {
        int i = idx / Q;
        int q = idx - i * Q;
        float di = dinv[i];
        float w  = di * di;
        float4 h = ((const float4*)H)[idx];
        float4 bb = ((const float4*)b)[q];
        float4 o;
        o.x = h.x * w + bb.x;  o.y = h.y * w + bb.y;
        o.z = h.z * w + bb.z;  o.w = h.w * w + bb.w;
        ((float4*)out)[idx] = o;
    }
}

// ------------------------- edge scatter-add --------------------------------
// out[dst][f..f+3] += H[src][f..f+3] * dinv[src]*dinv[dst]
// One thread per (edge, 4 features): float4 gather, 4 L2-resident atomics.
template<int F>
__global__ void gcn_scatter(const long long* __restrict__ ei,
                            const float* __restrict__ H,
                            const float* __restrict__ dinv,
                            float* __restrict__ out) {
    constexpr int Q = F / 4;
    int idx = blockIdx.x * blockDim.x + threadIdx.x;     // over E*F/4
    int e = idx / Q;
    if (e < GCN_E) {
        int q = idx - e * Q;
        int s = (int)ei[e];
        int d = (int)ei[GCN_E + e];
        float w = dinv[s] * dinv[d];
        float4 h = ((const float4*)(H + (size_t)s * F))[q];
        float* o = out + (size_t)d * F + q * 4;
        atomicAdd(o + 0, h.x * w);
        atomicAdd(o + 1, h.y * w);
        atomicAdd(o + 2, h.z * w);
        atomicAdd(o + 3, h.w * w);
    }
}

// ---------------------------------------------------------------------------

extern "C" void kernel_launch(void* const* d_in, const int* in_sizes, int n_in,
                              void* d_out, int out_size, void* d_ws, size_t ws_size,
                              hipStream_t stream) {
    const float*     x  = (const float*)    d_in[0];
    const long long* ei = (const long long*)d_in[1];   // int64 [2, E]
    const float*     W1 = (const float*)    d_in[2];
    const float*     b1 = (const float*)    d_in[3];
    const float*     W2 = (const float*)    d_in[4];
    const float*     b2 = (const float*)    d_in[5];
    const float*     W3 = (const float*)    d_in[6];
    const float*     b3 = (const float*)    d_in[7];
    float* out = (float*)d_out;

    // workspace layout: dinv[N] | bufA[N*128] | bufB[N*128]
    float* dinv = (float*)d_ws;
    float* bufA = dinv + GCN_N;
    float* bufB = bufA + (size_t)GCN_N * D_HID;

    const int T  = 256;
    const int gN = (GCN_N + T - 1) / T;
    const int gE = (GCN_E + T - 1) / T;

    // --- degree / dinv ---
    gcn_init_deg     <<<gN, T, 0, stream>>>(dinv, GCN_N);
    gcn_accum_deg    <<<gE, T, 0, stream>>>(ei, dinv, GCN_E);
    gcn_finalize_dinv<<<gN, T, 0, stream>>>(dinv, GCN_N);

    const int mTiles = GCN_N / 16;                         // 3125 (exact)
    const int gNQ128 = (GCN_N * (D_HID / 4) + T - 1) / T;
    const int gNQ64  = (GCN_N * (D_OUT / 4) + T - 1) / T;
    const int gEQ128 = (GCN_E * (D_HID / 4) + T - 1) / T;
    const int gEQ64  = (GCN_E * (D_OUT / 4) + T - 1) / T;

    // --- layer 1: x @ W1 -> bufA ; aggregate -> bufB ---
    gcn_gemm_wmma<D_IN, D_HID><<<mTiles, (D_HID/16)*32, 0, stream>>>(x, W1, bufA);
    gcn_self_bias<D_HID><<<gNQ128, T, 0, stream>>>(bufA, dinv, b1, bufB, GCN_N);
    gcn_scatter  <D_HID><<<gEQ128, T, 0, stream>>>(ei, bufA, dinv, bufB);

    // --- layer 2: bufB @ W2 -> bufA ; aggregate -> bufB ---
    gcn_gemm_wmma<D_HID, D_HID><<<mTiles, (D_HID/16)*32, 0, stream>>>(bufB, W2, bufA);
    gcn_self_bias<D_HID><<<gNQ128, T, 0, stream>>>(bufA, dinv, b2, bufB, GCN_N);
    gcn_scatter  <D_HID><<<gEQ128, T, 0, stream>>>(ei, bufA, dinv, bufB);

    // --- layer 3: bufB @ W3 -> bufA(N x 64) ; aggregate -> d_out ---
    gcn_gemm_wmma<D_HID, D_OUT><<<mTiles, (D_OUT/16)*32, 0, stream>>>(bufB, W3, bufA);
    gcn_self_bias<D_OUT><<<gNQ64, T, 0, stream>>>(bufA, dinv, b3, out, GCN_N);
    gcn_scatter  <D_OUT><<<gEQ64, T, 0, stream>>>(ei, bufA, dinv, out);
}